// AngularMultiCenterEmotionBall_49280454754357
// MI455X (gfx1250) — compile-verified
//
#include <hip/hip_runtime.h>
#include <hip/hip_bf16.h>
#include <math.h>

// ---------------------------------------------------------------------------
// AngularMultiCenterEmotionBall loss, MI455X (gfx1250, wave32, WMMA + async)
//
// B=131072, D=256, C=8, K=2 -> CK=16 centers.
// Memory-bound: one pass over z (128 MiB) @ 23.3 TB/s ~= 5.8 us floor.
// Data path: GLOBAL_LOAD_ASYNC_TO_LDS_B128 double-buffered z chunks (full
// cacheline consumption, ASYNCcnt-tracked prefetch) -> conflict-free
// ds_load_b64 fragments -> V_WMMA_F32_16X16X4_F32 (64 per 16-row tile).
// ---------------------------------------------------------------------------

#define B_N   131072
#define D_N   256
#define C_N   8
#define K_N   2
#define CK_N  16

#define TAU_INV     10.0f    // 1 / 0.1
#define MARGIN_OV   0.3f
#define MARGIN_DIV  0.8f

#define WAVES_PER_BLOCK 8
#define BLOCKS_MAIN     (B_N / (16 * WAVES_PER_BLOCK))   // 1024

#define CHUNK   32                  // K columns staged per async chunk
#define NCHUNK  (D_N / CHUNK)       // 8
#define ZROW    36                  // padded LDS row stride (floats): 16B-aligned
                                    // rows, 16 distinct bank pairs for b64 reads
#define CNROW   260                 // padded cn stride: bank shift 4, 8B aligned

typedef float v2f __attribute__((ext_vector_type(2)));
typedef float v8f __attribute__((ext_vector_type(8)));

// CDNA5 async copy: per-lane 16B global -> LDS, tracked by ASYNCcnt.
#define ASYNC_LOAD_B128(ldsoff, gptr)                                         \
  asm volatile("global_load_async_to_lds_b128 %0, %1, off"                    \
               :: "v"(ldsoff), "v"(gptr) : "memory")
#define WAIT_ASYNCCNT(n)                                                      \
  asm volatile("s_wait_asynccnt %0" :: "i"(n) : "memory")

// ws layout (floats):
//   [0,    4096)  c_norm  [16][256]
//   [4096, 5120)  per-block intra partial sums (BLOCKS_MAIN)

// ---------------------------------------------------------------------------
// Kernel A: normalize the 16 centers into workspace.
// ---------------------------------------------------------------------------
__global__ __launch_bounds__(256)
void emoball_prep_centers(const float* __restrict__ cb, float* __restrict__ cn) {
  __shared__ float ssq[CK_N][16];
  __shared__ float inv[CK_N];
  const int t = threadIdx.x;
  const int n = t >> 4;        // center 0..15
  const int j = t & 15;        // 16-way split of D

  float s = 0.0f;
  #pragma unroll
  for (int d = 0; d < 16; ++d) {
    float v = cb[n * D_N + j * 16 + d];
    s += v * v;
  }
  ssq[n][j] = s;
  __syncthreads();
  if (j == 0) {
    float acc = 0.0f;
    #pragma unroll
    for (int k = 0; k < 16; ++k) acc += ssq[n][k];
    inv[n] = 1.0f / fmaxf(sqrtf(acc), 1e-12f);
  }
  __syncthreads();
  const float iv = inv[n];
  #pragma unroll
  for (int d = 0; d < 16; ++d) {
    const int idx = n * D_N + j * 16 + d;
    cn[idx] = cb[idx] * iv;
  }
}

// ---------------------------------------------------------------------------
// Kernel B: L_intra partial sums. One wave = one 16-row tile of z.
// ---------------------------------------------------------------------------
__global__ __launch_bounds__(256)
void emoball_intra(const float* __restrict__ z,
                   const int*   __restrict__ labels,
                   const float* __restrict__ rel,
                   const float* __restrict__ radii,
                   const float* __restrict__ cn_g,
                   float*       __restrict__ partials) {
  __shared__ float cn_s[CK_N * CNROW];                        // ~16.6 KB
  __shared__ float ztile[WAVES_PER_BLOCK][2][16 * ZROW];      // ~36 KB, dbl buf
  __shared__ float wsum[WAVES_PER_BLOCK];

  const int t    = threadIdx.x;
  const int wave = t >> 5;
  const int lane = t & 31;
  const int half = lane >> 4;   // 0: K%4 in {0,1}, 1: K%4 in {2,3}
  const int l15  = lane & 15;   // row (A) / column (B,D) index

  const int rowbase = (blockIdx.x * WAVES_PER_BLOCK + wave) * 16;

  // Async-issue one 16-row x 32-col chunk of z into ztile[wave][buf].
  // 4 x B128: instruction k covers rows 4k..4k+3, each row = one full 128B run.
  auto issue = [&](int chunk, int buf) {
    const float* gbase = z + (size_t)rowbase * D_N + chunk * CHUNK;
    const unsigned lbase = (unsigned)(uintptr_t)&ztile[wave][buf][0];
    #pragma unroll
    for (int k = 0; k < 4; ++k) {
      const int idx = k * 32 + lane;
      const int r   = idx >> 3;       // 0..15
      const int seg = idx & 7;        // 16B segment within the 128B row chunk
      const unsigned loff = lbase + (unsigned)((r * ZROW + seg * 4) * 4);
      const unsigned long long g =
          (unsigned long long)(gbase + (size_t)r * D_N + seg * 4);
      ASYNC_LOAD_B128(loff, g);
    }
  };

  v8f   acc   = {};
  float ssq_l = 0.0f;
  const float* crow = cn_s + l15 * CNROW + 2 * half;

  // Consume 8 K-steps (WMMA K=4) from a staged chunk.
  auto process = [&](int chunk, int buf) {
    const float* zbuf = &ztile[wave][buf][0] + l15 * ZROW + 2 * half;
    #pragma unroll
    for (int ii = 0; ii < 8; ++ii) {
      v2f a = *(const v2f*)(zbuf + 4 * ii);                 // ds_load_b64
      v2f b = *(const v2f*)(crow + chunk * CHUNK + 4 * ii); // ds_load_b64
      ssq_l += a.x * a.x + a.y * a.y;                       // row |z|^2
      acc = __builtin_amdgcn_wmma_f32_16x16x4_f32(
          false, a, false, b, (short)0, acc, false, false);
    }
  };

  issue(0, 0);                               // prefetch overlaps cn staging

  for (int i = t; i < CK_N * D_N; i += 256)  // stage normalized centers
    cn_s[(i >> 8) * CNROW + (i & 255)] = cn_g[i];
  __syncthreads();

  for (int j = 0; j < NCHUNK - 1; ++j) {
    issue(j + 1, (j & 1) ^ 1);
    WAIT_ASYNCCNT(4);                        // chunk j landed (in-order done)
    process(j, j & 1);
  }
  WAIT_ASYNCCNT(0);
  process(NCHUNK - 1, (NCHUNK - 1) & 1);

  // inverse norm of row (rowbase + l15); lanes L and L+16 hold halves
  const float rowss = ssq_l + __shfl_xor(ssq_l, 16, 32);
  const float invn  = 1.0f / fmaxf(sqrtf(rowss), 1e-12f);

  // clipped radius for center n = l15
  const float rn = fminf(fmaxf(fabsf(radii[l15]), 0.05f), 1.0f);

  float intra = 0.0f;
  #pragma unroll
  for (int v = 0; v < 8; ++v) {
    const int   m     = v + 8 * half;                   // D row for this lane
    const float inv_m = __shfl(invn, m, 32);            // broadcast row norm
    const float simv  = acc[v] * inv_m;                 // normalized cosine
    const int   lab   = labels[rowbase + m];
    // masked softmax over this row's own-class columns (n>>1 == lab)
    const float e   = ((l15 >> 1) == lab) ? __expf(simv * TAU_INV) : 0.0f;
    float sw  = e;
    float swd = e * (1.0f - simv);
    float swr = e * rn;
    #pragma unroll
    for (int off = 8; off >= 1; off >>= 1) {            // 16-lane half reduce
      sw  += __shfl_xor(sw,  off, 32);
      swd += __shfl_xor(swd, off, 32);
      swr += __shfl_xor(swr, off, 32);
    }
    const float dist_w = swd / sw;
    const float r_w    = swr / sw;
    const float val    = rel[rowbase + m] * fmaxf(dist_w - r_w, 0.0f);
    if (l15 == 0) intra += val;                         // one lane per row
  }
  intra += __shfl_xor(intra, 16, 32);
  if (lane == 0) wsum[wave] = intra;
  __syncthreads();
  if (t == 0) {
    float s = 0.0f;
    #pragma unroll
    for (int w = 0; w < WAVES_PER_BLOCK; ++w) s += wsum[w];
    partials[blockIdx.x] = s;
  }
}

// ---------------------------------------------------------------------------
// Kernel C: fold partials, compute overlap/diversity on the 16x16 Gram, emit
// the final scalar.
// ---------------------------------------------------------------------------
__global__ __launch_bounds__(256)
void emoball_finalize(const float* __restrict__ cn,
                      const float* __restrict__ partials,
                      float*       __restrict__ out) {
  __shared__ float r0[256], r1[256], r2[256];
  const int t = threadIdx.x;

  float s = 0.0f;
  for (int i = t; i < BLOCKS_MAIN; i += 256) s += partials[i];

  const int p = t >> 4;
  const int q = t & 15;
  float dotpq = 0.0f;
  for (int d = 0; d < D_N; ++d) dotpq += cn[p * D_N + d] * cn[q * D_N + d];

  const bool same_class = ((p >> 1) == (q >> 1));
  const float ov = (!same_class) ? fmaxf(dotpq - MARGIN_OV, 0.0f) : 0.0f;
  const float dv = (same_class && (p < q)) ? fmaxf(dotpq - MARGIN_DIV, 0.0f) : 0.0f;

  r0[t] = s; r1[t] = ov; r2[t] = dv;
  __syncthreads();
  for (int st = 128; st > 0; st >>= 1) {
    if (t < st) { r0[t] += r0[t + st]; r1[t] += r1[t + st]; r2[t] += r2[t + st]; }
    __syncthreads();
  }
  if (t == 0) {
    const float L_intra = r0[0] / (float)B_N;
    const float mask_n  = (float)(CK_N * CK_N - C_N * K_N * K_N);     // 224
    const float L_ov    = r1[0] / (mask_n + 1e-6f);
    const float L_div   = r2[0] / (float)(C_N * K_N * (K_N - 1) / 2); // 8
    out[0] = 1.0f * L_intra + 0.5f * L_ov + 0.5f * L_div;
  }
}

// ---------------------------------------------------------------------------
extern "C" void kernel_launch(void* const* d_in, const int* in_sizes, int n_in,
                              void* d_out, int out_size, void* d_ws, size_t ws_size,
                              hipStream_t stream) {
  (void)in_sizes; (void)n_in; (void)out_size; (void)ws_size;
  const float* z      = (const float*)d_in[0];
  const int*   labels = (const int*)  d_in[1];
  const float* rel    = (const float*)d_in[2];
  const float* cb     = (const float*)d_in[3];
  const float* radii  = (const float*)d_in[4];

  float* ws       = (float*)d_ws;
  float* cn       = ws;             // 4096 floats
  float* partials = ws + 4096;      // BLOCKS_MAIN floats

  emoball_prep_centers<<<1, 256, 0, stream>>>(cb, cn);
  emoball_intra<<<BLOCKS_MAIN, 256, 0, stream>>>(z, labels, rel, radii, cn, partials);
  emoball_finalize<<<1, 256, 0, stream>>>(cn, partials, (float*)d_out);
}